// HierarchicalGCN_88905823027257
// MI455X (gfx1250) — compile-verified
//
#include <hip/hip_runtime.h>
#include <hip/hip_bf16.h>

typedef __attribute__((ext_vector_type(2))) float v2f;
typedef __attribute__((ext_vector_type(8))) float v8f;

#define NDIM 128
#define SX_PITCH 132   // pad LDS rows to dodge bank conflicts

// ---------------- degree / normalization ----------------

__global__ __launch_bounds__(256) void k_init_cnt(int* __restrict__ cnt, int n) {
    int i = blockIdx.x * 256 + threadIdx.x;
    if (i < n) cnt[i] = 1;  // self-loop contributes 1 to degree
}

__global__ __launch_bounds__(256) void k_count(const int* __restrict__ dst,
                                               int* __restrict__ cnt, int nE) {
    int i = blockIdx.x * 256 + threadIdx.x;
    if (i < nE) atomicAdd(&cnt[dst[i]], 1);
}

__global__ __launch_bounds__(256) void k_dinv(const int* __restrict__ cnt,
                                              float* __restrict__ dinv, int n) {
    int i = blockIdx.x * 256 + threadIdx.x;
    if (i < n) dinv[i] = rsqrtf((float)cnt[i]);
}

// ---------------- GEMM: H[m0:m0+16, 0:128] = (relu?)X @ W via fp32 WMMA ----------------
// Block = 256 threads = 8 waves. Wave w owns output columns [16w, 16w+16).
// Guard-free: every block handles a full 16-row tile (tail rows go to k_gemm_tail).

template <bool RELU>
__global__ __launch_bounds__(256) void k_gemm128(const float* __restrict__ X,
                                                 const float* __restrict__ W,
                                                 float* __restrict__ H) {
    __shared__ float sX[16 * SX_PITCH];
    const int tid = threadIdx.x;
    const int m0  = blockIdx.x * 16;

    // cooperative load of 16x128 X tile (with optional fused ReLU)
    for (int i = tid; i < 16 * NDIM; i += 256) {
        int r = i >> 7, c = i & 127;
        float v = X[(size_t)(m0 + r) * NDIM + c];
        if (RELU) v = fmaxf(v, 0.0f);
        sX[r * SX_PITCH + c] = v;
    }
    __syncthreads();

    const int wave = tid >> 5;
    const int lane = tid & 31;
    const int lo   = lane & 15;    // row (A) / col (B) / col (C)
    const int hi   = lane >> 4;    // selects K pair in A/B, +8 rows in C
    const int n0   = wave * 16;

    v8f acc = {};
    #pragma unroll 8
    for (int kb = 0; kb < NDIM; kb += 4) {
        const int k = kb + 2 * hi;
        v2f a, b;
        // A 16x4: lane lo row m, VGPR0 = K=k, VGPR1 = K=k+1
        a.x = sX[lo * SX_PITCH + k];
        a.y = sX[lo * SX_PITCH + k + 1];
        // B 4x16: lane lo column n0+lo, VGPR0 = K=k, VGPR1 = K=k+1
        b.x = W[(size_t)k * NDIM + n0 + lo];
        b.y = W[(size_t)(k + 1) * NDIM + n0 + lo];
        acc = __builtin_amdgcn_wmma_f32_16x16x4_f32(
            /*neg_a=*/false, a, /*neg_b=*/false, b,
            /*c_mod=*/(short)0, acc, /*reuse_a=*/false, /*reuse_b=*/false);
    }

    // C/D layout: VGPR v holds row m0 + v + 8*hi, column n0 + lo.
    // Single base + 8 stores at 512B-immediate offsets -> one s_clause.
    float* __restrict__ out = H + (size_t)(m0 + 8 * hi) * NDIM + n0 + lo;
    #pragma unroll
    for (int v = 0; v < 8; ++v)
        out[(size_t)v * NDIM] = acc[v];
}

// scalar tail for nNodes % 16 rows (not launched when nNodes is a multiple of 16)
__global__ __launch_bounds__(128) void k_gemm_tail(const float* __restrict__ X,
                                                   const float* __restrict__ W,
                                                   float* __restrict__ H,
                                                   int relu_in, int row0, int nNodes) {
    int row = row0 + blockIdx.x;
    if (row >= nNodes) return;
    int col = threadIdx.x;
    float s = 0.0f;
    for (int k = 0; k < NDIM; ++k) {
        float v = X[(size_t)row * NDIM + k];
        if (relu_in) v = fmaxf(v, 0.0f);
        s += v * W[(size_t)k * NDIM + col];
    }
    H[(size_t)row * NDIM + col] = s;
}

// ---------------- self-loop + bias init: out = h * dinv^2 + b ----------------

__global__ __launch_bounds__(256) void k_agginit(const float* __restrict__ H,
                                                 const float* __restrict__ dinv,
                                                 const float* __restrict__ bias,
                                                 float* __restrict__ out, int total) {
    int i = blockIdx.x * 256 + threadIdx.x;
    if (i < total) {
        int n = i >> 7, j = i & 127;
        float di = dinv[n];
        out[i] = H[i] * di * di + bias[j];
    }
}

// ---------------- edge scatter: one wave per edge, lane -> float4 chunk ----------------

__global__ __launch_bounds__(256) void k_scatter(const float* __restrict__ H,
                                                 const int* __restrict__ src,
                                                 const int* __restrict__ dst,
                                                 const float* __restrict__ dinv,
                                                 float* __restrict__ out, int nE) {
    int e = blockIdx.x * 8 + (threadIdx.x >> 5);
    if (e >= nE) return;
    int lane = threadIdx.x & 31;
    int s = src[e], d = dst[e];
    float norm = dinv[s] * dinv[d];
    float4 v = ((const float4*)(H + (size_t)s * NDIM))[lane];
    float* o = out + (size_t)d * NDIM + lane * 4;
    unsafeAtomicAdd(o + 0, v.x * norm);
    unsafeAtomicAdd(o + 1, v.y * norm);
    unsafeAtomicAdd(o + 2, v.z * norm);
    unsafeAtomicAdd(o + 3, v.w * norm);
}

// ---------------- pooling ----------------

__global__ __launch_bounds__(256) void k_zero_pool(float* __restrict__ psum,
                                                   int* __restrict__ gcount) {
    int i = blockIdx.x * 256 + threadIdx.x;
    if (i < 128 * 128) psum[i] = 0.0f;
    if (i < 128) gcount[i] = 0;
}

__global__ __launch_bounds__(256) void k_pool(const float* __restrict__ H,
                                              const int* __restrict__ batch,
                                              float* __restrict__ psum,
                                              int* __restrict__ gcount, int nNodes) {
    int nid = blockIdx.x * 8 + (threadIdx.x >> 5);
    if (nid >= nNodes) return;
    int lane = threadIdx.x & 31;
    int g = batch[nid];
    float4 v = ((const float4*)(H + (size_t)nid * NDIM))[lane];
    float* p = psum + (size_t)g * NDIM + lane * 4;
    unsafeAtomicAdd(p + 0, v.x);
    unsafeAtomicAdd(p + 1, v.y);
    unsafeAtomicAdd(p + 2, v.z);
    unsafeAtomicAdd(p + 3, v.w);
    if (lane == 0) atomicAdd(&gcount[g], 1);
}

// ---------------- fused mean + final linear: out[g,o] = (sum/cnt) @ Wlin + blin ----------------

__global__ __launch_bounds__(64) void k_final(const float* __restrict__ psum,
                                              const int* __restrict__ gcount,
                                              const float* __restrict__ Wlin,
                                              const float* __restrict__ blin,
                                              float* __restrict__ out) {
    int g = blockIdx.x;
    int o = threadIdx.x;
    float inv = 1.0f / fmaxf((float)gcount[g], 1.0f);
    float s = 0.0f;
    #pragma unroll 8
    for (int k = 0; k < NDIM; ++k)
        s += psum[(size_t)g * NDIM + k] * Wlin[k * 64 + o];
    out[g * 64 + o] = s * inv + blin[o];
}

// ---------------- host launch ----------------

extern "C" void kernel_launch(void* const* d_in, const int* in_sizes, int n_in,
                              void* d_out, int out_size, void* d_ws, size_t ws_size,
                              hipStream_t stream) {
    const float* x     = (const float*)d_in[0];
    const int*   ei    = (const int*)d_in[1];
    const int*   batch = (const int*)d_in[2];
    const float* W1 = (const float*)d_in[3];
    const float* b1 = (const float*)d_in[4];
    const float* W2 = (const float*)d_in[5];
    const float* b2 = (const float*)d_in[6];
    const float* W3 = (const float*)d_in[7];
    const float* b3 = (const float*)d_in[8];
    const float* Wlin = (const float*)d_in[9];
    const float* blin = (const float*)d_in[10];

    const int nNodes = in_sizes[0] / NDIM;   // 50000
    const int nE     = in_sizes[1] / 2;      // 1600000
    const int* src = ei;
    const int* dst = ei + nE;

    char* ws = (char*)d_ws;
    size_t off = 0;
    auto salloc = [&](size_t bytes) -> void* {
        void* p = ws + off;
        off += (bytes + 255) & ~(size_t)255;
        return p;
    };
    int*   cnt    = (int*)  salloc((size_t)nNodes * 4);
    float* dinv   = (float*)salloc((size_t)nNodes * 4);
    float* hbuf   = (float*)salloc((size_t)nNodes * NDIM * 4);
    float* aggbuf = (float*)salloc((size_t)nNodes * NDIM * 4);
    float* psum   = (float*)salloc(128 * 128 * 4);
    int*   gcount = (int*)  salloc(128 * 4);

    const int total   = nNodes * NDIM;
    const int nb      = (nNodes + 255) / 256;
    const int eb      = (nE + 255) / 256;
    const int tb      = (total + 255) / 256;
    const int mfull   = nNodes / 16;         // guard-free full tiles
    const int mrem    = nNodes - mfull * 16; // tail rows (0 for 50000)
    const int eblocks = (nE + 7) / 8;
    const int pblocks = (nNodes + 7) / 8;

    auto gemm = [&](const float* in, const float* W, float* out, bool relu) {
        if (mfull > 0) {
            if (relu) k_gemm128<true><<<mfull, 256, 0, stream>>>(in, W, out);
            else      k_gemm128<false><<<mfull, 256, 0, stream>>>(in, W, out);
        }
        if (mrem > 0)
            k_gemm_tail<<<mrem, 128, 0, stream>>>(in, W, out, relu ? 1 : 0,
                                                  mfull * 16, nNodes);
    };

    // degrees & symmetric normalization
    k_init_cnt<<<nb, 256, 0, stream>>>(cnt, nNodes);
    k_count  <<<eb, 256, 0, stream>>>(dst, cnt, nE);
    k_dinv   <<<nb, 256, 0, stream>>>(cnt, dinv, nNodes);

    // layer 1
    gemm(x, W1, hbuf, false);
    k_agginit<<<tb, 256, 0, stream>>>(hbuf, dinv, b1, aggbuf, total);
    k_scatter<<<eblocks, 256, 0, stream>>>(hbuf, src, dst, dinv, aggbuf, nE);

    // layer 2 (ReLU of layer-1 output fused into GEMM load)
    gemm(aggbuf, W2, hbuf, true);
    k_agginit<<<tb, 256, 0, stream>>>(hbuf, dinv, b2, aggbuf, total);
    k_scatter<<<eblocks, 256, 0, stream>>>(hbuf, src, dst, dinv, aggbuf, nE);

    // layer 3 (ReLU of layer-2 output fused into GEMM load; no ReLU after)
    gemm(aggbuf, W3, hbuf, true);
    k_agginit<<<tb, 256, 0, stream>>>(hbuf, dinv, b3, aggbuf, total);
    k_scatter<<<eblocks, 256, 0, stream>>>(hbuf, src, dst, dinv, aggbuf, nE);

    // global mean pool + final linear
    k_zero_pool<<<(128 * 128 + 255) / 256, 256, 0, stream>>>(psum, gcount);
    k_pool<<<pblocks, 256, 0, stream>>>(aggbuf, batch, psum, gcount, nNodes);
    k_final<<<128, 64, 0, stream>>>(psum, gcount, Wlin, blin, (float*)d_out);
}